// HIPABlock_25967372271691
// MI455X (gfx1250) — compile-verified
//
#include <hip/hip_runtime.h>
#include <hip/hip_bf16.h>
#include <math.h>

#define Bb 8
#define Cc 256
#define Hh 224
#define Ww 224
#define OUTC 256
#define GRID 4
#define KH 56           // 224/4
#define CELLS 16
#define NROWS (Bb*CELLS)   // 128
#define PLANE (Hh*Ww)      // 50176
#define BPLANE (Cc*Hh*Ww)  // 12845056

typedef float v2f __attribute__((ext_vector_type(2)));
typedef float v8f __attribute__((ext_vector_type(8)));

// ---------------------------------------------------------------------------
// Kernel 1: 4x4 grid max pool. One wave per (b,c,cell); 56x56 = 3136 elems.
// coarse layout: [b][c][cell]  (index = (b*C + c)*16 + cell)
// ---------------------------------------------------------------------------
__global__ void pool_kernel(const float* __restrict__ x, float* __restrict__ coarse) {
    int wave = (blockIdx.x * blockDim.x + threadIdx.x) >> 5;   // 0..32767
    int lane = threadIdx.x & 31;
    int cell = wave & 15;
    int bc   = wave >> 4;          // b*C + c  == plane index
    int gy = cell >> 2, gx = cell & 3;
    const float* base = x + (long long)bc * PLANE + gy * KH * Ww + gx * KH;
    float mx = -INFINITY;
    for (int e = lane; e < KH * KH; e += 32) {
        int r = e / KH;
        int col = e - r * KH;
        mx = fmaxf(mx, base[r * Ww + col]);
    }
    #pragma unroll
    for (int off = 16; off > 0; off >>= 1)
        mx = fmaxf(mx, __shfl_xor(mx, off, 32));
    if (lane == 0) coarse[wave] = mx;
}

// ---------------------------------------------------------------------------
// Kernel 2: per (b,cell): LayerNorm over C + importance = ||coarse||_2.
// One block (256 thr == C) per row (b*16+cell).  normed: [row][c] row-major.
// ---------------------------------------------------------------------------
__global__ void stats_kernel(const float* __restrict__ coarse,
                             const float* __restrict__ gamma,
                             const float* __restrict__ beta,
                             float* __restrict__ normed,
                             float* __restrict__ importance) {
    __shared__ float ssum[256];
    __shared__ float ssq[256];
    int row  = blockIdx.x;         // b*16 + cell
    int b    = row >> 4;
    int cell = row & 15;
    int c    = threadIdx.x;
    float v = coarse[(b * Cc + c) * CELLS + cell];
    ssum[c] = v;
    ssq[c]  = v * v;
    __syncthreads();
    for (int s = 128; s > 0; s >>= 1) {
        if (c < s) { ssum[c] += ssum[c + s]; ssq[c] += ssq[c + s]; }
        __syncthreads();
    }
    float mean = ssum[0] * (1.0f / Cc);
    float var  = ssq[0] * (1.0f / Cc) - mean * mean;
    float rstd = rsqrtf(var + 1e-5f);
    normed[row * Cc + c] = (v - mean) * rstd * gamma[c] + beta[c];
    if (c == 0) importance[row] = sqrtf(ssq[0]);
}

// ---------------------------------------------------------------------------
// Kernel 3: per-batch mask / keep / coords. One wave per batch.
// ---------------------------------------------------------------------------
__global__ void mask_kernel(const float* __restrict__ importance,
                            const float* __restrict__ logit_th,
                            float* __restrict__ mask_ws,
                            float* __restrict__ out_keep,
                            float* __restrict__ out_coords) {
    int b = blockIdx.x;
    int lane = threadIdx.x;                 // 0..31
    float imp = importance[b * CELLS + (lane & 15)];  // lanes 16-31 duplicate
    float mn = imp, mx = imp;
    #pragma unroll
    for (int off = 16; off > 0; off >>= 1) {
        mn = fminf(mn, __shfl_xor(mn, off, 32));
        mx = fmaxf(mx, __shfl_xor(mx, off, 32));
    }
    float thresh = 1.0f / (1.0f + expf(-logit_th[2]));   // level NUM_LEVELS-1
    float imp01 = (imp - mn) / (mx - mn + 1e-8f);
    float hard = (imp01 >= thresh) ? 1.0f : 0.0f;
    float soft = 1.0f / (1.0f + expf(-10.0f * (imp01 - thresh)));
    float mval = fmaxf((hard - soft) + soft, 0.0f);      // STE forward, clamp 0
    if (lane < 16) {
        int row = b * CELLS + lane;
        mask_ws[row] = mval;
        float keepv = (mval > 0.5f) ? 1.0f : 0.0f;
        out_keep[row] = keepv;
        int ix = lane & 3, iy = lane >> 2;
        float cx = (ix + 0.5f) * 0.25f;
        float cy = (iy + 0.5f) * 0.25f;
        float* oc = out_coords + row * 4;
        oc[0] = cx * keepv; oc[1] = cy * keepv;
        oc[2] = 0.25f * keepv; oc[3] = 0.25f * keepv;
    }
}

// ---------------------------------------------------------------------------
// Kernel 4: proj = LN(coarse) @ W + bias, gated by keep.  (128x256)·(256x256)
// V_WMMA_F32_16X16X4_F32, one wave per 16x16 output tile, K-loop of 64 steps.
// A 16x4 f32 frag: lanes0-15 -> K0/K1, lanes16-31 -> K2/K3 (VGPR0/VGPR1).
// B 4x16 frag: same K split, N = lane&15.  D: VGPR i -> row i+8*half.
// ---------------------------------------------------------------------------
__global__ void proj_wmma_kernel(const float* __restrict__ A,    // [128][256]
                                 const float* __restrict__ Wp,   // [256 k][256 n]
                                 const float* __restrict__ bias, // [256]
                                 const float* __restrict__ mask_ws, // [128]
                                 float* __restrict__ out_feats) {  // [128][256]
    int wave = (blockIdx.x * blockDim.x + threadIdx.x) >> 5;  // 0..127
    int lane = threadIdx.x & 31;
    int mt = wave >> 4;        // 0..7
    int nt = wave & 15;        // 0..15
    int m0 = mt * 16, n0 = nt * 16;
    int half = lane >> 4;
    int lm   = lane & 15;
    const float* arow = A  + (m0 + lm) * Cc;   // row of A for this lane
    const float* bcol = Wp + n0 + lm;          // column of W for this lane
    v8f acc = {};
    for (int k0 = 0; k0 < Cc; k0 += 4) {
        int ka = k0 + 2 * half;
        v2f a; a.x = arow[ka];            a.y = arow[ka + 1];
        v2f bf; bf.x = bcol[ka * OUTC];   bf.y = bcol[(ka + 1) * OUTC];
        acc = __builtin_amdgcn_wmma_f32_16x16x4_f32(
            false, a, false, bf, (short)0, acc, false, false);
    }
    int col = n0 + lm;
    float bv = bias[col];
    #pragma unroll
    for (int i = 0; i < 8; ++i) {
        int row = m0 + i + 8 * half;
        float keepv = (mask_ws[row] > 0.5f) ? 1.0f : 0.0f;
        out_feats[row * OUTC + col] = (acc[i] + bv) * keepv;
    }
}

// ---------------------------------------------------------------------------
// Kernel 5: fused bilinear (half-pixel, edge-clamp) 4x4->224x224 upsample of
// the mask, multiplied into x.  float4 vectorized; one block spans one batch.
// ---------------------------------------------------------------------------
__global__ void upsample_mul_kernel(const float* __restrict__ x,
                                    const float* __restrict__ mask_ws,
                                    float* __restrict__ out) {
    __shared__ float m[16];
    int base_blk = blockIdx.x * 1024;           // 256 thr * 4 floats
    int b = base_blk / BPLANE;                  // uniform across block
    if (threadIdx.x < 16) m[threadIdx.x] = mask_ws[b * CELLS + threadIdx.x];
    __syncthreads();
    int idx = base_blk + (threadIdx.x << 2);
    int hw = idx % PLANE;
    int h  = hw / Ww;
    int w0 = hw - h * Ww;                       // multiple of 4, no row cross

    const float scale = (float)GRID / (float)Hh;  // 1/56
    float sy = (h + 0.5f) * scale - 0.5f;
    float fy = floorf(sy);
    float ty = sy - fy;
    int iy = (int)fy;
    int y0 = min(max(iy, 0), 3);
    int y1 = min(max(iy + 1, 0), 3);

    float4 xv = *(const float4*)(x + idx);
    float r[4];
    float xin[4] = {xv.x, xv.y, xv.z, xv.w};
    #pragma unroll
    for (int j = 0; j < 4; ++j) {
        float sxp = (w0 + j + 0.5f) * scale - 0.5f;
        float fx = floorf(sxp);
        float tx = sxp - fx;
        int ix = (int)fx;
        int x0 = min(max(ix, 0), 3);
        int x1 = min(max(ix + 1, 0), 3);
        float top = m[(y0 << 2) + x0] * (1.0f - tx) + m[(y0 << 2) + x1] * tx;
        float bot = m[(y1 << 2) + x0] * (1.0f - tx) + m[(y1 << 2) + x1] * tx;
        float mv  = top * (1.0f - ty) + bot * ty;
        r[j] = xin[j] * mv;
    }
    float4 ov = {r[0], r[1], r[2], r[3]};
    *(float4*)(out + idx) = ov;
}

// ---------------------------------------------------------------------------
extern "C" void kernel_launch(void* const* d_in, const int* in_sizes, int n_in,
                              void* d_out, int out_size, void* d_ws, size_t ws_size,
                              hipStream_t stream) {
    const float* x        = (const float*)d_in[0];
    const float* logit_th = (const float*)d_in[1];
    const float* ln_gamma = (const float*)d_in[2];
    const float* ln_beta  = (const float*)d_in[3];
    const float* w_proj   = (const float*)d_in[4];   // (C, OUT) row-major
    const float* b_proj   = (const float*)d_in[5];

    float* out = (float*)d_out;
    const int N_SPARSE = Bb * Cc * Hh * Ww;          // 102760448
    float* out_sparse = out;
    float* out_feats  = out + N_SPARSE;              // 128*256
    float* out_coords = out_feats + NROWS * OUTC;    // 128*4
    float* out_keep   = out_coords + NROWS * 4;      // 128

    float* ws         = (float*)d_ws;
    float* coarse     = ws;                          // 32768
    float* normed     = coarse + Bb * Cc * CELLS;    // 32768
    float* importance = normed + NROWS * Cc;         // 128
    float* mask_ws    = importance + NROWS;          // 128

    // 1) grid max pool: 32768 waves, 8 waves/block
    pool_kernel<<<(Bb * Cc * CELLS) / 8, 256, 0, stream>>>(x, coarse);

    // 2) LN + importance: one block per (b,cell)
    stats_kernel<<<NROWS, 256, 0, stream>>>(coarse, ln_gamma, ln_beta,
                                            normed, importance);

    // 3) mask / keep / coords: one wave per batch
    mask_kernel<<<Bb, 32, 0, stream>>>(importance, logit_th,
                                       mask_ws, out_keep, out_coords);

    // 4) WMMA projection: 128 waves = 16 blocks x 256 threads
    proj_wmma_kernel<<<16, 256, 0, stream>>>(normed, w_proj, b_proj,
                                             mask_ws, out_feats);

    // 5) fused upsample * x -> out_sparse (float4, 1024 elems/block)
    upsample_mul_kernel<<<N_SPARSE / 1024, 256, 0, stream>>>(x, mask_ws,
                                                             out_sparse);
}